// NTXentLoss_84645215470199
// MI455X (gfx1250) — compile-verified
//
#include <hip/hip_runtime.h>
#include <hip/hip_bf16.h>
#include <math.h>

typedef float v2f __attribute__((ext_vector_type(2)));
typedef float v8f __attribute__((ext_vector_type(8)));
typedef unsigned int u32;
typedef u32 u32x4 __attribute__((ext_vector_type(4)));
typedef int  i32x4 __attribute__((ext_vector_type(4)));
typedef int  i32x8 __attribute__((ext_vector_type(8)));

#define N_ROWS 4096
#define TWO_N  8192
#define DIM    128
#define INV_T  (1.0f / 0.07f)
#define EPSN   1e-12f
#define LDS_STRIDE 132   // 128 + 4 pad floats; 132 % 64 = 4 -> conflict-free ds reads
#define NTILES (TWO_N / 16)

__device__ __forceinline__ const float* row_ptr(const float* __restrict__ zi,
                                                const float* __restrict__ zj, int g) {
    return (g < N_ROWS) ? (zi + (size_t)g * DIM) : (zj + (size_t)(g - N_ROWS) * DIM);
}

// ---------------------------------------------------------------------------
// Kernel 1: per-row reciprocal norms + positive-pair logits.
// ---------------------------------------------------------------------------
__global__ void ntxent_rownorm(const float* __restrict__ zi, const float* __restrict__ zj,
                               float* __restrict__ rn, float* __restrict__ pos) {
    const int i = blockIdx.x;
    const int t = threadIdx.x;            // 128 threads, one per D element
    const float a = zi[(size_t)i * DIM + t];
    const float b = zj[(size_t)i * DIM + t];
    float na = a * a, nb = b * b, dab = a * b;
    #pragma unroll
    for (int m = 16; m >= 1; m >>= 1) {   // wave32 butterfly
        na  += __shfl_xor(na,  m, 32);
        nb  += __shfl_xor(nb,  m, 32);
        dab += __shfl_xor(dab, m, 32);
    }
    __shared__ float red[3][4];
    const int w = t >> 5;
    if ((t & 31) == 0) { red[0][w] = na; red[1][w] = nb; red[2][w] = dab; }
    __syncthreads();
    if (t == 0) {
        na  = red[0][0] + red[0][1] + red[0][2] + red[0][3];
        nb  = red[1][0] + red[1][1] + red[1][2] + red[1][3];
        dab = red[2][0] + red[2][1] + red[2][2] + red[2][3];
        const float ri = 1.0f / fmaxf(sqrtf(na), EPSN);
        const float rj = 1.0f / fmaxf(sqrtf(nb), EPSN);
        rn[i]          = ri;
        rn[i + N_ROWS] = rj;
        pos[i]         = dab * ri * rj * INV_T;
    }
}

// ---------------------------------------------------------------------------
// Kernel 2: fused sim = z z^T / T  +  row-wise online logsumexp (diag excluded).
// 64 blocks x 256 threads (8 waves). Wave w owns rows [blk*128 + w*16, +16).
// A operand resident in VGPRs; B (column) tile DMA'd into LDS by the Tensor
// Data Mover, double-buffered, with TDM LDS-padding producing the 132-float
// row stride the ds reads expect. fp32 WMMA 16x16x4.
// ---------------------------------------------------------------------------
__global__ __launch_bounds__(256) void ntxent_sim_lse(
        const float* __restrict__ zi, const float* __restrict__ zj,
        const float* __restrict__ rn, float* __restrict__ lse_out) {
    __shared__ float tileB[2][16 * LDS_STRIDE];

    const int tid  = threadIdx.x;
    const int wave = tid >> 5;
    const int lane = tid & 31;
    const int half = lane >> 4;           // K-half selector for A/B operands
    const int l15  = lane & 15;
    const int row0 = blockIdx.x * 128 + wave * 16;

    // TDM issue: 2D tile, 16 rows x 128 f32, row stride 128 f32 in memory,
    // +4 DWORD LDS pad every 128 DWORDs (pad_interval code 6, pad_amount code 3).
    auto issue_tdm = [&](int ct, int buf) {
        const int g = ct * 16;            // first column-row of the tile
        const float* base = (g < N_ROWS) ? (zi + (size_t)g * DIM)
                                         : (zj + (size_t)(g - N_ROWS) * DIM);
        const unsigned long long ga = (unsigned long long)(uintptr_t)base;
        const u32 lds = (u32)(unsigned long long)(uintptr_t)&tileB[buf][0];
        // D# group 0: count=1 | lds_addr | global_addr[56:0] | type=2
        u32x4 g0 = { 1u,                                   // [1:0] count = 1
                     lds,                                  // [63:32] lds_addr
                     (u32)ga,                              // [95:64] global_addr lo
                     ((u32)(ga >> 32) & 0x01FFFFFFu) | 0x80000000u }; // [120:96] hi, [127:126] type=2
        // D# group 1:
        //  d0: data_size=2 (4B) | pad_enable | pad_interval=6 (128 DW) | pad_amount=3 (4 DW)
        //  d1: tensor_dim0[15:0]=128 << 16
        //  d2: tensor_dim1[15:0]=4096 << 16
        //  d3: tile_dim0=128 << 16
        //  d4: tile_dim1=16
        //  d5: tensor_dim0_stride=128
        i32x8 g1 = { (int)0x07920000,
                     (int)(128u << 16),
                     (int)(4096u << 16),
                     (int)(128u << 16),
                     16, 128, 0, 0 };
        i32x4 zz4 = { 0, 0, 0, 0 };
        i32x8 zz8 = { 0, 0, 0, 0, 0, 0, 0, 0 };
        __builtin_amdgcn_tensor_load_to_lds(g0, g1, zz4, zz4, zz8, 0);
    };

    // --- A operand: this wave's 16 rows x 128 K, WMMA 32-bit A layout:
    //     lanes 0-15 hold M=lane (K = 4k+0,4k+1), lanes 16-31 hold M=lane-16 (K = 4k+2,4k+3)
    const float* ap = row_ptr(zi, zj, row0 + l15) + 2 * half;
    v2f areg[32];
    #pragma unroll
    for (int kk = 0; kk < 32; ++kk)
        areg[kk] = *(const v2f*)(ap + 4 * kk);

    // --- per-C-row reciprocal norms (C layout: VGPR r -> rows r / r+8 by lane half)
    float rrow[8];
    #pragma unroll
    for (int r = 0; r < 8; ++r)
        rrow[r] = rn[row0 + r + 8 * half];

    float mrun[8], srun[8];
    #pragma unroll
    for (int r = 0; r < 8; ++r) { mrun[r] = -INFINITY; srun[r] = 0.0f; }

    if (wave == 0) issue_tdm(0, 0);       // prologue: DMA tile 0 into buffer 0

    for (int ct = 0; ct < NTILES; ++ct) {
        const int buf = ct & 1;

        if (wave == 0) __builtin_amdgcn_s_wait_tensorcnt(0);
        __syncthreads();                  // tile `ct` now visible to all waves
        if (wave == 0 && (ct + 1) < NTILES)
            issue_tdm(ct + 1, buf ^ 1);   // overlap next DMA with compute

        // 16x16 raw-dot tile: 32 x V_WMMA_F32_16X16X4_F32 over K=128.
        v8f c = {0.f, 0.f, 0.f, 0.f, 0.f, 0.f, 0.f, 0.f};
        const float* bp = &tileB[buf][l15 * LDS_STRIDE + 2 * half]; // B layout == A layout of col rows
        #pragma unroll
        for (int kk = 0; kk < 32; ++kk) {
            v2f b = *(const v2f*)(bp + 4 * kk);
            c = __builtin_amdgcn_wmma_f32_16x16x4_f32(
                    /*neg_a=*/false, areg[kk], /*neg_b=*/false, b,
                    /*c_mod=*/(short)0, c, /*reuse_a=*/false, /*reuse_b=*/false);
        }

        // Online LSE update per row (16-lane-half butterfly reductions).
        const int   col0 = ct * 16;
        const float rcol = rn[col0 + l15];
        const int   jcol = col0 + l15;
        #pragma unroll
        for (int r = 0; r < 8; ++r) {
            const int irow = row0 + r + 8 * half;
            float x = c[r] * rrow[r] * rcol * INV_T;
            x = (irow == jcol) ? -INFINITY : x;   // exclude diagonal

            float mt = x;
            mt = fmaxf(mt, __shfl_xor(mt, 1, 32));
            mt = fmaxf(mt, __shfl_xor(mt, 2, 32));
            mt = fmaxf(mt, __shfl_xor(mt, 4, 32));
            mt = fmaxf(mt, __shfl_xor(mt, 8, 32));
            const float mnew = fmaxf(mrun[r], mt);

            float ex = __expf(x - mnew);          // exp(-inf - m) == 0
            ex += __shfl_xor(ex, 1, 32);
            ex += __shfl_xor(ex, 2, 32);
            ex += __shfl_xor(ex, 4, 32);
            ex += __shfl_xor(ex, 8, 32);

            srun[r] = srun[r] * __expf(mrun[r] - mnew) + ex;
            mrun[r] = mnew;
        }
    }

    #pragma unroll
    for (int r = 0; r < 8; ++r)
        if (l15 == 0)
            lse_out[row0 + r + 8 * half] = mrun[r] + logf(srun[r]);
}

// ---------------------------------------------------------------------------
// Kernel 3: loss = (sum(lse) - 2*sum(pos_half)) / 2N
// ---------------------------------------------------------------------------
__global__ void ntxent_finalize(const float* __restrict__ lse, const float* __restrict__ pos,
                                float* __restrict__ out) {
    const int t = threadIdx.x;   // 256 threads
    float acc = 0.0f;
    for (int i = t; i < TWO_N; i += 256)  acc += lse[i];
    for (int i = t; i < N_ROWS; i += 256) acc -= 2.0f * pos[i];
    #pragma unroll
    for (int m = 16; m >= 1; m >>= 1) acc += __shfl_xor(acc, m, 32);
    __shared__ float red[8];
    if ((t & 31) == 0) red[t >> 5] = acc;
    __syncthreads();
    if (t == 0) {
        float s = 0.0f;
        #pragma unroll
        for (int w = 0; w < 8; ++w) s += red[w];
        out[0] = s / (float)TWO_N;
    }
}

extern "C" void kernel_launch(void* const* d_in, const int* in_sizes, int n_in,
                              void* d_out, int out_size, void* d_ws, size_t ws_size,
                              hipStream_t stream) {
    const float* zi = (const float*)d_in[0];
    const float* zj = (const float*)d_in[1];
    float* ws  = (float*)d_ws;
    float* rn  = ws;                 // [8192]
    float* lse = ws + TWO_N;         // [8192]
    float* pos = ws + 2 * TWO_N;     // [4096]

    ntxent_rownorm<<<N_ROWS, 128, 0, stream>>>(zi, zj, rn, pos);
    ntxent_sim_lse<<<TWO_N / 128, 256, 0, stream>>>(zi, zj, rn, lse);
    ntxent_finalize<<<1, 256, 0, stream>>>(lse, pos, (float*)d_out);
}